// Depth_AE_52424370815669
// MI455X (gfx1250) — compile-verified
//
#include <hip/hip_runtime.h>
#include <math.h>

// ---------------- problem constants ----------------
#define BB 28
#define CCH 2816
#define FF (CCH*16*16)          // 720896
#define NTOT (BB*FF)            // 20185088

typedef __attribute__((ext_vector_type(2))) float v2f;
typedef __attribute__((ext_vector_type(8))) float v8f;

__device__ __forceinline__ float silu_f(float v) { return v / (1.0f + expf(-v)); }

// =========================================================
// deterministic global sum / sumsq (double partials, no atomics)
// =========================================================
__global__ void k_reduce1(const float* __restrict__ x,
                          double* __restrict__ psum, double* __restrict__ psq) {
    __shared__ double ss[256];
    __shared__ double sq[256];
    double s = 0.0, q = 0.0;
    for (long i = (long)blockIdx.x * 256 + threadIdx.x; i < (long)NTOT; i += 1024L * 256L) {
        double v = (double)x[i];
        s += v; q += v * v;
    }
    ss[threadIdx.x] = s; sq[threadIdx.x] = q;
    __syncthreads();
    for (int o = 128; o > 0; o >>= 1) {
        if ((int)threadIdx.x < o) {
            ss[threadIdx.x] += ss[threadIdx.x + o];
            sq[threadIdx.x] += sq[threadIdx.x + o];
        }
        __syncthreads();
    }
    if (threadIdx.x == 0) { psum[blockIdx.x] = ss[0]; psq[blockIdx.x] = sq[0]; }
}

__global__ void k_reduce2(const double* __restrict__ psum, const double* __restrict__ psq,
                          float* __restrict__ stdp) {
    __shared__ double ss[256];
    __shared__ double sq[256];
    double s = 0.0, q = 0.0;
    for (int i = threadIdx.x; i < 1024; i += 256) { s += psum[i]; q += psq[i]; }
    ss[threadIdx.x] = s; sq[threadIdx.x] = q;
    __syncthreads();
    for (int o = 128; o > 0; o >>= 1) {
        if ((int)threadIdx.x < o) {
            ss[threadIdx.x] += ss[threadIdx.x + o];
            sq[threadIdx.x] += sq[threadIdx.x + o];
        }
        __syncthreads();
    }
    if (threadIdx.x == 0) {
        double n = (double)NTOT;
        double var = (sq[0] - ss[0] * ss[0] / n) / (n - 1.0);  // ddof=1
        stdp[0] = (float)sqrt(var > 0.0 ? var : 0.0);
    }
}

// =========================================================
// per-feature quantiles -> 1/bw (rank-count selection, registers only)
// =========================================================
__global__ void k_bw(const float* __restrict__ x, const float* __restrict__ stdp,
                     float* __restrict__ invbw) {
    int f = blockIdx.x * 256 + threadIdx.x;
    if (f >= FF) return;
    float v[BB];
#pragma unroll
    for (int s = 0; s < BB; s++) v[s] = x[(long)s * FF + f];
    float q6 = 0.f, q7 = 0.f, q20 = 0.f, q21 = 0.f;
#pragma unroll
    for (int i = 0; i < BB; i++) {
        int r = 0;
#pragma unroll
        for (int j = 0; j < BB; j++)
            r += (v[j] < v[i]) || ((v[j] == v[i]) && (j < i));
        if (r == 6)  q6  = v[i];
        if (r == 7)  q7  = v[i];
        if (r == 20) q20 = v[i];
        if (r == 21) q21 = v[i];
    }
    float q25 = q6  + 0.75f * (q7  - q6);    // pos 0.25*27 = 6.75
    float q75 = q20 + 0.25f * (q21 - q20);   // pos 0.75*27 = 20.25
    float iqr = q75 - q25;
    float sd = stdp[0];
    float bw = 0.35829645349814747f * fminf(sd, iqr * (1.0f / 1.34f)); // 0.9*100^-0.2
    invbw[f] = 1.0f / bw;
}

// =========================================================
// CDF mean
// =========================================================
__global__ void k_cdf(const float* __restrict__ x, const float* __restrict__ invbw,
                      float* __restrict__ out) {
    int f = blockIdx.x * 256 + threadIdx.x;
    if (f >= FF) return;
    int b = blockIdx.y;
    float xb = x[(long)b * FF + f];
    float ib = invbw[f];
    float acc = 0.0f;
#pragma unroll
    for (int s = 0; s < BB; s++) {
        float z = (xb - x[(long)s * FF + f]) * ib;
        acc += erff(z * 0.70710678118654752f);
    }
    out[(long)b * FF + f] = 0.5f + acc * (0.5f / (float)BB);
}

// =========================================================
// zero-fill (for padded intermediates)
// =========================================================
__global__ void k_zero(float* __restrict__ p, long n) {
    long i = (long)blockIdx.x * 256 + threadIdx.x;
    if (i < n) p[i] = 0.0f;
}

// =========================================================
// conv1: M=4732 (28*13*13), N=44 (pad 48), K=45056 (2816*16)
// one wave -> 16x16 tile; 8 waves/block share one N-tile, B staged in LDS
// =========================================================
#define M1 4732
#define K1 45056
#define NT1 3
#define MB1 37   // 296/8

__global__ void k_conv1(const float* __restrict__ A, const float* __restrict__ w1,
                        const float* __restrict__ bias, float* __restrict__ out) {
    __shared__ float Bs[16 * 68];              // [n_local*68 + k_local], bank-padded
    const int tid  = threadIdx.x;
    const int lane = tid & 31;
    const int wv   = tid >> 5;
    const int nt   = blockIdx.x / MB1;
    const int mt   = (blockIdx.x % MB1) * 8 + wv;
    const int nl   = lane & 15;
    const int n    = nt * 16 + nl;
    const bool nok = (n < 44);
    int m = mt * 16 + nl; if (m > M1 - 1) m = M1 - 1;   // clamp: pad rows load real data
    const int bI = m / 169; int rI = m % 169; const int oy = rI / 13, ox = rI % 13;
    const long abase = (long)bI * CCH * 256 + oy * 16 + ox;
    const int khalf = (lane >> 4) * 2;                   // 0 | 2

    // cooperative-load mapping: thread -> (row n_l, 16-byte segment)
    const int cl_n = tid >> 4, cl_s = tid & 15;
    int cln = nt * 16 + cl_n; if (cln > 43) cln = 43;
    const float4* wrow = (const float4*)(w1 + (long)cln * K1);  // row base is 16B aligned

    v8f c = {0.f,0.f,0.f,0.f,0.f,0.f,0.f,0.f};
    for (int c0 = 0; c0 < CCH; c0 += 4) {               // K-chunk = 4 channels = 64 k
        const int f4 = c0 * 4;                          // float4 index of chunk start
        float4 seg = wrow[f4 + cl_s];
        *(float4*)&Bs[cl_n * 68 + cl_s * 4] = seg;
        if (c0 + 4 < CCH)
            __builtin_prefetch((const void*)(wrow + f4 + 16 + cl_s), 0, 1);
        __syncthreads();
#pragma unroll
        for (int cl = 0; cl < 4; ++cl) {
#pragma unroll
            for (int r0 = 0; r0 < 16; r0 += 4) {
                const int rr = r0 + khalf;               // {0,2,...,14}
                const int ky = rr >> 2, kx = rr & 3;     // kx in {0,2} -> pair contiguous
                const long ao = abase + (long)(c0 + cl) * 256 + ky * 16 + kx;
                v2f av; av[0] = A[ao]; av[1] = A[ao + 1];
                v2f bv = *(const v2f*)&Bs[nl * 68 + cl * 16 + rr];
                c = __builtin_amdgcn_wmma_f32_16x16x4_f32(false, av, false, bv,
                                                          (short)0, c, false, false);
            }
        }
        __syncthreads();
    }
    float bn = nok ? bias[n] : 0.0f;
#pragma unroll
    for (int r = 0; r < 8; r++) {
        int mo = mt * 16 + r + ((lane >= 16) ? 8 : 0);
        if (mo < M1 && nok) {
            int bo = mo / 169, pos = mo % 169, yy = pos / 13, xx = pos % 13;
            out[(((long)bo * 44 + n) * 13 + yy) * 13 + xx] = silu_f(c[r] + bn);
        }
    }
}

// =========================================================
// pool: min/avg/max 3x3 (28,44,13,13) -> (28,132,11,11)
// =========================================================
__global__ void k_pool(const float* __restrict__ h1, float* __restrict__ out) {
    int idx = blockIdx.x * 256 + threadIdx.x;
    if (idx >= 28 * 44 * 121) return;
    int xx = idx % 11; int t = idx / 11;
    int yy = t % 11;   t /= 11;
    int ch = t % 44;   int b = t / 44;
    const float* p = h1 + (((long)b * 44 + ch) * 13 + yy) * 13 + xx;
    float mn = INFINITY, mx = -INFINITY, sm = 0.0f;
#pragma unroll
    for (int dy = 0; dy < 3; dy++)
#pragma unroll
        for (int dx = 0; dx < 3; dx++) {
            float v = p[dy * 13 + dx];
            mn = fminf(mn, v); mx = fmaxf(mx, v); sm += v;
        }
    long base = (long)b * 132 * 121 + yy * 11 + xx;
    out[base + (long)ch * 121]        = mn;
    out[base + (long)(44 + ch) * 121] = sm * (1.0f / 9.0f);
    out[base + (long)(88 + ch) * 121] = mx;
}

// =========================================================
// conv2: grouped 1x1 (3 x 44->44) + SiLU, writes into zero-padded (28,132,19,19)
// =========================================================
__global__ void k_conv2(const float* __restrict__ hp, const float* __restrict__ w2,
                        const float* __restrict__ bias, float* __restrict__ outp) {
    int idx = blockIdx.x * 256 + threadIdx.x;      // 28*132*121
    if (idx >= 28 * 132 * 121) return;
    int px = idx % 11; int t = idx / 11;
    int py = t % 11;   t /= 11;
    int oc = t % 132;  int b = t / 132;
    int g = oc / 44;
    const float* w  = w2 + (long)oc * 44;
    const float* in = hp + ((long)b * 132 + g * 44) * 121 + py * 11 + px;
    float acc = bias[oc];
#pragma unroll 4
    for (int ci = 0; ci < 44; ci++) acc += w[ci] * in[(long)ci * 121];
    outp[(((long)b * 132 + oc) * 19 + py + 4) * 19 + px + 4] = silu_f(acc);
}

// =========================================================
// convT3: M=6300 (28*15*15), N=352, K=3300 (132*25); input zero-padded 19x19
// out written into zero-padded (28,352,17,17) for convT4
// =========================================================
#define M3 6300
#define NT3 22
#define MB3 50   // ceil(394/8)

__global__ void k_conv3(const float* __restrict__ Ap, const float* __restrict__ w3,
                        const float* __restrict__ bias, float* __restrict__ outp) {
    __shared__ float Bs[16 * 104];                 // [n_local*104 + k_local] (k_local<100)
    const int tid  = threadIdx.x;
    const int lane = tid & 31;
    const int wv   = tid >> 5;
    const int nt   = blockIdx.x / MB3;
    const int mt   = (blockIdx.x % MB3) * 8 + wv;
    const int nl   = lane & 15;
    const int n    = nt * 16 + nl;                 // N exact multiple of 16
    int m = mt * 16 + nl; if (m > M3 - 1) m = M3 - 1;
    const int bI = m / 225; int rI = m % 225; const int oy = rI / 15, ox = rI % 15;
    const long abase = (long)bI * 132 * 361 + (oy + 4) * 19 + (ox + 4);
    const int khalf = (lane >> 4) * 2;

    v8f c = {0.f,0.f,0.f,0.f,0.f,0.f,0.f,0.f};
    for (int ci0 = 0; ci0 < 132; ci0 += 4) {       // K-chunk = 4 ci = 100 k
        for (int idx = tid; idx < 1600; idx += 256) {
            int rk = idx % 25, rest = idx / 25;
            int n_l = rest & 15, ci_l = rest >> 4;
            Bs[n_l * 104 + ci_l * 25 + rk] =
                w3[((long)(ci0 + ci_l) * 352 + nt * 16 + n_l) * 25 + rk];
        }
        __syncthreads();
#pragma unroll 5
        for (int s = 0; s < 25; ++s) {
            int ka = s * 4 + khalf;
            int ci = ka / 25, rk = ka % 25, ky = rk / 5, kx = rk % 5;
            float a0 = Ap[abase + (long)(ci0 + ci) * 361 - ky * 19 - kx];
            int kb = ka + 1;
            int ci2 = kb / 25, rk2 = kb % 25, ky2 = rk2 / 5, kx2 = rk2 % 5;
            float a1 = Ap[abase + (long)(ci0 + ci2) * 361 - ky2 * 19 - kx2];
            v2f av; av[0] = a0; av[1] = a1;
            v2f bv = *(const v2f*)&Bs[nl * 104 + ka];
            c = __builtin_amdgcn_wmma_f32_16x16x4_f32(false, av, false, bv,
                                                      (short)0, c, false, false);
        }
        __syncthreads();
    }
    float bn = bias[n];
#pragma unroll
    for (int r = 0; r < 8; r++) {
        int mo = mt * 16 + r + ((lane >= 16) ? 8 : 0);
        if (mo < M3) {
            int bo = mo / 225, pos = mo % 225, yy = pos / 15, xx = pos % 15;
            outp[(((long)bo * 352 + n) * 17 + yy + 1) * 17 + xx + 1] = silu_f(c[r] + bn);
        }
    }
}

// =========================================================
// convT4: M=7168 (28*16*16, exact), N=2816 (exact), K=1408 (352*4)
// input zero-padded (28,352,17,17); leaky ReLU -> d_out
// =========================================================
#define NT4 176
#define MB4 56   // 448/8 exact

__global__ void k_conv4(const float* __restrict__ Ap, const float* __restrict__ w4,
                        const float* __restrict__ bias, float* __restrict__ out) {
    __shared__ float Bs[16 * 68];
    const int tid  = threadIdx.x;
    const int lane = tid & 31;
    const int wv   = tid >> 5;
    const int nt   = blockIdx.x / MB4;
    const int mt   = (blockIdx.x % MB4) * 8 + wv;
    const int nl   = lane & 15;
    const int n    = nt * 16 + nl;
    const int m    = mt * 16 + nl;
    const int bI = m >> 8, oy = (m >> 4) & 15, ox = m & 15;
    const long abase = (long)bI * 352 * 289 + (oy + 1) * 17 + (ox + 1);
    const int kyoff = (lane >= 16) ? 17 : 0;       // khalf=2 -> ky=1
    const int khalf = (lane >> 4) * 2;

    const int cl_ci = tid >> 4, cl_n = tid & 15;
    const float4* w4v = (const float4*)w4;         // w4[(ci*2816+n)*4 .. +3] = float4[ci*2816+n]

    v8f c = {0.f,0.f,0.f,0.f,0.f,0.f,0.f,0.f};
    for (int ci0 = 0; ci0 < 352; ci0 += 16) {      // K-chunk = 16 ci = 64 k
        float4 seg = w4v[(long)(ci0 + cl_ci) * CCH + nt * 16 + cl_n];
        *(float4*)&Bs[cl_n * 68 + cl_ci * 4] = seg;
        __syncthreads();
#pragma unroll
        for (int s = 0; s < 16; ++s) {
            const long ao = abase + (long)(ci0 + s) * 289 - kyoff;
            v2f av; av[0] = Ap[ao]; av[1] = Ap[ao - 1];   // kx = 0, 1
            v2f bv = *(const v2f*)&Bs[nl * 68 + s * 4 + khalf];
            c = __builtin_amdgcn_wmma_f32_16x16x4_f32(false, av, false, bv,
                                                      (short)0, c, false, false);
        }
        __syncthreads();
    }
    float bn = bias[n];
#pragma unroll
    for (int r = 0; r < 8; r++) {
        int mo = mt * 16 + r + ((lane >= 16) ? 8 : 0);
        int bo = mo >> 8, pos = mo & 255, yy = pos >> 4, xx = pos & 15;
        float v = c[r] + bn;
        v = (v > 0.0f) ? v : 0.01f * v;            // leaky_relu(0.01)
        out[(((long)bo * CCH + n) * 16 + yy) * 16 + xx] = v;
    }
}

// =========================================================
// launcher
// =========================================================
extern "C" void kernel_launch(void* const* d_in, const int* in_sizes, int n_in,
                              void* d_out, int out_size, void* d_ws, size_t ws_size,
                              hipStream_t stream) {
    const float* x  = (const float*)d_in[0];
    const float* w1 = (const float*)d_in[1];
    const float* b1 = (const float*)d_in[2];
    const float* w2 = (const float*)d_in[3];
    const float* b2 = (const float*)d_in[4];
    const float* w3 = (const float*)d_in[5];
    const float* b3 = (const float*)d_in[6];
    const float* w4 = (const float*)d_in[7];
    const float* b4 = (const float*)d_in[8];
    float* out = (float*)d_out;

    char* ws = (char*)d_ws;
    double* psum = (double*)ws;                       // 1024 doubles
    double* psq  = psum + 1024;                       // 1024 doubles
    float*  stdp = (float*)(psum + 2048);
    float*  hqn   = (float*)(ws + 32768);             // 20,185,088
    float*  invbw = hqn + (long)NTOT;                 //    720,896
    float*  h1    = invbw + FF;                       // 28*44*169   = 208,208
    float*  hpool = h1 + 28 * 44 * 169;               // 28*132*121  = 447,216
    float*  h2pad = hpool + 28 * 132 * 121;           // 28*132*19*19 = 1,334,256
    float*  h3pad = h2pad + 28 * 132 * 361;           // 28*352*17*17 = 2,848,384
    const long padN = 28L * 132 * 361 + 28L * 352 * 289; // both pads contiguous

    // quantnorm
    k_reduce1<<<1024, 256, 0, stream>>>(x, psum, psq);
    k_reduce2<<<1, 256, 0, stream>>>(psum, psq, stdp);
    k_bw<<<(FF + 255) / 256, 256, 0, stream>>>(x, stdp, invbw);
    dim3 gc((FF + 255) / 256, BB);
    k_cdf<<<gc, 256, 0, stream>>>(x, invbw, hqn);

    // zero the padded intermediates (pad halo must be 0 every call)
    k_zero<<<(int)((padN + 255) / 256), 256, 0, stream>>>(h2pad, padN);

    // conv stack
    k_conv1<<<NT1 * MB1, 256, 0, stream>>>(hqn, w1, b1, h1);          // 111 blocks
    k_pool<<<(28 * 44 * 121 + 255) / 256, 256, 0, stream>>>(h1, hpool);
    k_conv2<<<(28 * 132 * 121 + 255) / 256, 256, 0, stream>>>(hpool, w2, b2, h2pad);
    k_conv3<<<NT3 * MB3, 256, 0, stream>>>(h2pad, w3, b3, h3pad);     // 1100 blocks
    k_conv4<<<NT4 * MB4, 256, 0, stream>>>(h3pad, w4, b4, out);       // 9856 blocks
}